// Net_40733469835604
// MI455X (gfx1250) — compile-verified
//
#include <hip/hip_runtime.h>
#include <math.h>

typedef __attribute__((ext_vector_type(2))) float v2f;
typedef __attribute__((ext_vector_type(8))) float v8f;

static inline int cdiv(int a, int b) { return (a + b - 1) / b; }

// ---------------------------------------------------------------------------
// order-preserving float <-> uint encoding for atomicMax-based segment_max
// ---------------------------------------------------------------------------
__device__ __forceinline__ unsigned fenc(float f) {
    unsigned u = __float_as_uint(f);
    return (u & 0x80000000u) ? ~u : (u | 0x80000000u);
}
__device__ __forceinline__ float fdec(unsigned u) {
    return __uint_as_float((u & 0x80000000u) ? (u & 0x7fffffffu) : ~u);
}

__global__ void pool_scatter_kernel(const float* __restrict__ x, const int* __restrict__ cl,
                                    unsigned* __restrict__ out, int Nin, int C) {
    int tid = blockIdx.x * blockDim.x + threadIdx.x;
    if (tid >= Nin * C) return;
    int n = tid / C;
    int c = tid - n * C;
    atomicMax(&out[(size_t)cl[n] * C + c], fenc(x[tid]));
}

__global__ void pool_decode_kernel(unsigned* __restrict__ buf, int total) {
    int tid = blockIdx.x * blockDim.x + threadIdx.x;
    if (tid >= total) return;
    buf[tid] = __float_as_uint(fdec(buf[tid]));
}

// ---------------------------------------------------------------------------
// degree histogram: deg[dst] += 1
// ---------------------------------------------------------------------------
__global__ void deg_kernel(const int* __restrict__ ei, float* __restrict__ deg, int E) {
    int tid = blockIdx.x * blockDim.x + threadIdx.x;
    if (tid >= E) return;
    atomicAdd(&deg[ei[E + tid]], 1.0f);
}

// ---------------------------------------------------------------------------
// spline-basis scatter:  Zp[dst, kidx*Cin + c] += basis * h[src,c] / deg[dst]
// ---------------------------------------------------------------------------
__global__ void spline_scatter_kernel(const float* __restrict__ h,
                                      const int* __restrict__ ei,
                                      const float* __restrict__ pseudo,
                                      const float* __restrict__ deg,
                                      float* __restrict__ Zp,
                                      int E, int Cin, int Kdp, int chunks) {
    int tid = blockIdx.x * blockDim.x + threadIdx.x;
    if (tid >= E * chunks) return;
    int e = tid / chunks;
    int q = tid - e * chunks;
    int src = ei[e];
    int dst = ei[E + e];

    float f[3];
    int i0[3];
#pragma unroll
    for (int d = 0; d < 3; ++d) {
        float v  = pseudo[e * 3 + d] * 4.0f;            // (K-1) = 4
        float fl = fminf(fmaxf(floorf(v), 0.0f), 3.0f); // clip to [0, K-2]
        i0[d] = (int)fl;
        f[d]  = v - fl;
    }
    float inv = 1.0f / fmaxf(deg[dst], 1.0f);

    float basis[8];
    int   kidx[8];
    const int mult[3] = {25, 5, 1};
#pragma unroll
    for (int b = 0; b < 8; ++b) {
        float w = 1.0f;
        int   k = 0;
#pragma unroll
        for (int d = 0; d < 3; ++d) {
            int bit = (b >> d) & 1;
            w *= bit ? f[d] : (1.0f - f[d]);
            k += (i0[d] + bit) * mult[d];
        }
        basis[b] = w;
        kidx[b]  = k;
    }

    size_t base = (size_t)dst * Kdp;
    if (Cin == 1) {
        float xv = h[src] * inv;
#pragma unroll
        for (int b = 0; b < 8; ++b)
            atomicAdd(&Zp[base + kidx[b]], basis[b] * xv);
    } else {
        int c0 = q * 4;
        const float4 xv = *(const float4*)(h + (size_t)src * Cin + c0);
#pragma unroll
        for (int b = 0; b < 8; ++b) {
            float s = basis[b] * inv;
            float* p = &Zp[base + (size_t)kidx[b] * Cin + c0];
            atomicAdd(p + 0, s * xv.x);
            atomicAdd(p + 1, s * xv.y);
            atomicAdd(p + 2, s * xv.z);
            atomicAdd(p + 3, s * xv.w);
        }
    }
}

// copy h into the "root" tail columns of Zp (unscaled)
__global__ void tail_copy_kernel(const float* __restrict__ h, float* __restrict__ Zp,
                                 int N, int Cin, int Kdp) {
    int tid = blockIdx.x * blockDim.x + threadIdx.x;
    if (tid >= N * Cin) return;
    int n = tid / Cin;
    int c = tid - n * Cin;
    Zp[(size_t)n * Kdp + 125 * Cin + c] = h[tid];
}

// ---------------------------------------------------------------------------
// Packed-B layout for the fp32 WMMA B-operand:
//   element (k, n)  ->  PB[(k>>2)*4*N + n*4 + (k&3)]
// so lane L (n = L&15, khalf = L>>4) loads its (k0+2h, k0+2h+1) pair with one
// contiguous b64, coalesced across lanes.
// ---------------------------------------------------------------------------

// W_cat = [W ; root] packed  (padded k rows pre-zeroed by memset)
__global__ void wcat_pack_kernel(const float* __restrict__ W, const float* __restrict__ root,
                                 float* __restrict__ PB, int Cin, int Cout) {
    int total = 126 * Cin * Cout;
    int tid = blockIdx.x * blockDim.x + threadIdx.x;
    if (tid >= total) return;
    int k = tid / Cout;
    int n = tid - k * Cout;
    float v = (k < 125 * Cin) ? W[tid] : root[(size_t)(k - 125 * Cin) * Cout + n];
    PB[(size_t)(k >> 2) * 4 * Cout + (size_t)n * 4 + (k & 3)] = v;
}

// generic row-major [K,N] -> packed (for fc1_w)
__global__ void pack_b_kernel(const float* __restrict__ B, float* __restrict__ PB,
                              int K, int N) {
    int tid = blockIdx.x * blockDim.x + threadIdx.x;
    if (tid >= K * N) return;
    int k = tid / N;
    int n = tid - k * N;
    PB[(size_t)(k >> 2) * 4 * N + (size_t)n * 4 + (k & 3)] = B[tid];
}

// ---------------------------------------------------------------------------
// fp32 WMMA GEMM:  C[M,N] = A[M,Kd] @ B[Kd,N] + bias (+ELU)
// V_WMMA_F32_16X16X4_F32; one wave owns a 32 x (16*NT) tile (2 m-subtiles
// share every B load). Software-pipelined k-loop: next A/B preloaded before
// the current WMMAs issue.
// A layout (ISA 7.12.2, 32-bit 16x4): lane = M + 16*(K/2), vgpr = K%2
// C/D layout: vgpr r holds D[r + 8*(lane/16)][lane%16]
// ---------------------------------------------------------------------------
template <int NT>
__global__ __launch_bounds__(256) void wmma_gemm_kernel(
    const float* __restrict__ A, int lda,
    const float* __restrict__ PB, int N,
    const float* __restrict__ bias,
    float* __restrict__ C, int ldc,
    int MtGroups, int NtGroups, int Kd, int do_elu) {
    int wave = blockIdx.x * (blockDim.x >> 5) + (threadIdx.x >> 5);
    int lane = threadIdx.x & 31;
    int mt = wave / NtGroups;
    int ng = wave - mt * NtGroups;
    if (mt >= MtGroups) return;

    int m0 = mt * 32;
    int n0 = ng * (16 * NT);
    int r  = lane & 15;   // row of A-subtile / col of B- and D-tiles
    int hf = lane >> 4;   // k-half selector / D row-half selector

    v8f acc0[NT] = {};
    v8f acc1[NT] = {};

    const float* pA0 = A + (size_t)(m0 + r) * lda + 2 * hf;
    const float* pA1 = pA0 + (size_t)16 * lda;
    const float* pB  = PB + (size_t)(n0 + r) * 4 + 2 * hf;
    const size_t bstep = (size_t)4 * N;

    v2f a0 = *(const v2f*)pA0;
    v2f a1 = *(const v2f*)pA1;
    v2f bb[NT];
#pragma unroll
    for (int t = 0; t < NT; ++t) bb[t] = *(const v2f*)(pB + (size_t)t * 64);

    for (int k0 = 0; k0 + 4 < Kd; k0 += 4) {
        pA0 += 4;
        pA1 += 4;
        pB  += bstep;
        v2f a0n = *(const v2f*)pA0;
        v2f a1n = *(const v2f*)pA1;
        v2f bbn[NT];
#pragma unroll
        for (int t = 0; t < NT; ++t) bbn[t] = *(const v2f*)(pB + (size_t)t * 64);
#pragma unroll
        for (int t = 0; t < NT; ++t) {
            acc0[t] = __builtin_amdgcn_wmma_f32_16x16x4_f32(
                false, a0, false, bb[t], (short)0, acc0[t], false, false);
            acc1[t] = __builtin_amdgcn_wmma_f32_16x16x4_f32(
                false, a1, false, bb[t], (short)0, acc1[t], false, false);
        }
        a0 = a0n;
        a1 = a1n;
#pragma unroll
        for (int t = 0; t < NT; ++t) bb[t] = bbn[t];
    }
#pragma unroll
    for (int t = 0; t < NT; ++t) {
        acc0[t] = __builtin_amdgcn_wmma_f32_16x16x4_f32(
            false, a0, false, bb[t], (short)0, acc0[t], false, false);
        acc1[t] = __builtin_amdgcn_wmma_f32_16x16x4_f32(
            false, a1, false, bb[t], (short)0, acc1[t], false, false);
    }

#pragma unroll
    for (int t = 0; t < NT; ++t) {
        int n = n0 + t * 16 + r;
        float bv = bias[n];
#pragma unroll
        for (int rr = 0; rr < 8; ++rr) {
            float v0 = acc0[t][rr] + bv;
            float v1 = acc1[t][rr] + bv;
            if (do_elu) {
                v0 = (v0 > 0.0f) ? v0 : expm1f(v0);
                v1 = (v1 > 0.0f) ? v1 : expm1f(v1);
            }
            C[(size_t)(m0 + rr + 8 * hf) * ldc + n]      = v0;
            C[(size_t)(m0 + 16 + rr + 8 * hf) * ldc + n] = v1;
        }
    }
}

// ---------------------------------------------------------------------------
// FC2 (512 -> 10) + log_softmax;  one wave per row
// ---------------------------------------------------------------------------
__global__ void fc2_logsoftmax_kernel(const float* __restrict__ h, const float* __restrict__ w,
                                      const float* __restrict__ b, float* __restrict__ out) {
    int row  = blockIdx.x;
    int lane = threadIdx.x;
    __shared__ float sl[10];
    float acc = 0.0f;
    if (lane < 10) {
        for (int k = 0; k < 512; ++k) acc += h[row * 512 + k] * w[k * 10 + lane];
        acc += b[lane];
        sl[lane] = acc;
    }
    __syncthreads();
    if (lane < 10) {
        float mx = sl[0];
        for (int j = 1; j < 10; ++j) mx = fmaxf(mx, sl[j]);
        float s = 0.0f;
        for (int j = 0; j < 10; ++j) s += expf(sl[j] - mx);
        out[row * 10 + lane] = acc - mx - logf(s);
    }
}

// ---------------------------------------------------------------------------
// Host orchestration
// ---------------------------------------------------------------------------
extern "C" void kernel_launch(void* const* d_in, const int* in_sizes, int n_in,
                              void* d_out, int out_size, void* d_ws, size_t ws_size,
                              hipStream_t stream) {
    // inputs (setup_inputs order)
    const float* x        = (const float*)d_in[0];
    const int*   cluster0 = (const int*)d_in[1];
    const int*   ei1      = (const int*)d_in[2];
    const float* ps1      = (const float*)d_in[3];
    const int*   cluster1 = (const int*)d_in[4];
    const int*   ei2      = (const int*)d_in[5];
    const float* ps2      = (const float*)d_in[6];
    const int*   cluster2 = (const int*)d_in[7];
    const int*   ei3      = (const int*)d_in[8];
    const float* ps3      = (const float*)d_in[9];
    const int*   cluster3 = (const int*)d_in[10];
    const int*   ei4      = (const int*)d_in[11];
    const float* ps4      = (const float*)d_in[12];
    const int*   cluster4 = (const int*)d_in[13];
    const float* W1 = (const float*)d_in[14];
    const float* r1 = (const float*)d_in[15];
    const float* b1 = (const float*)d_in[16];
    const float* W2 = (const float*)d_in[17];
    const float* r2 = (const float*)d_in[18];
    const float* b2 = (const float*)d_in[19];
    const float* W3 = (const float*)d_in[20];
    const float* r3 = (const float*)d_in[21];
    const float* b3 = (const float*)d_in[22];
    const float* W4 = (const float*)d_in[23];
    const float* r4 = (const float*)d_in[24];
    const float* b4 = (const float*)d_in[25];
    const float* fc1_w = (const float*)d_in[26];
    const float* fc1_b = (const float*)d_in[27];
    const float* fc2_w = (const float*)d_in[28];
    const float* fc2_b = (const float*)d_in[29];
    (void)in_sizes; (void)n_in; (void)out_size;

    // workspace carve-up
    constexpr size_t Z_BYTES  = 16384ull * 4032ull * 4ull;  // 252 MiB (max over layers)
    constexpr size_t WC_BYTES = 16128ull * 256ull * 4ull;   // 15.75 MiB
    constexpr size_t P_BYTES  = 4ull << 20;
    constexpr size_t H_BYTES  = 4ull << 20;
    constexpr size_t D_BYTES  = 32768ull * 4ull;
    if (ws_size < Z_BYTES + WC_BYTES + P_BYTES + H_BYTES + D_BYTES) return;

    char*  wsb  = (char*)d_ws;
    float* Z    = (float*)(wsb);
    float* WC   = (float*)(wsb + Z_BYTES);
    float* PBUF = (float*)(wsb + Z_BYTES + WC_BYTES);
    float* HBUF = (float*)(wsb + Z_BYTES + WC_BYTES + P_BYTES);
    float* DEG  = (float*)(wsb + Z_BYTES + WC_BYTES + P_BYTES + H_BYTES);

    auto pool = [&](const float* src, const int* cl, int Nin, int Nout, int C, float* dst) {
        hipMemsetAsync(dst, 0, (size_t)Nout * C * sizeof(float), stream);
        pool_scatter_kernel<<<cdiv(Nin * C, 256), 256, 0, stream>>>(src, cl, (unsigned*)dst, Nin, C);
        pool_decode_kernel<<<cdiv(Nout * C, 256), 256, 0, stream>>>((unsigned*)dst, Nout * C);
    };

    auto conv = [&](const float* hin, int N, int Cin, int Cout,
                    const int* ei, const float* ps, int E,
                    const float* W, const float* root, const float* bias, float* hout) {
        int Kd  = 126 * Cin;
        int Kdp = (Kd + 3) & ~3;
        hipMemsetAsync(DEG, 0, (size_t)N * sizeof(float), stream);
        hipMemsetAsync(Z, 0, (size_t)N * Kdp * sizeof(float), stream);
        hipMemsetAsync(WC, 0, (size_t)Kdp * Cout * sizeof(float), stream);
        deg_kernel<<<cdiv(E, 256), 256, 0, stream>>>(ei, DEG, E);
        int chunks = (Cin == 1) ? 1 : Cin / 4;
        spline_scatter_kernel<<<cdiv(E * chunks, 256), 256, 0, stream>>>(
            hin, ei, ps, DEG, Z, E, Cin, Kdp, chunks);
        tail_copy_kernel<<<cdiv(N * Cin, 256), 256, 0, stream>>>(hin, Z, N, Cin, Kdp);
        wcat_pack_kernel<<<cdiv(126 * Cin * Cout, 256), 256, 0, stream>>>(W, root, WC, Cin, Cout);
        int MtG = N / 32;
        if (Cout == 32) {
            int NtG = 1;
            wmma_gemm_kernel<2><<<cdiv(MtG * NtG, 8), 256, 0, stream>>>(
                Z, Kdp, WC, Cout, bias, hout, Cout, MtG, NtG, Kdp, 1);
        } else {
            int NtG = Cout / 64;
            wmma_gemm_kernel<4><<<cdiv(MtG * NtG, 8), 256, 0, stream>>>(
                Z, Kdp, WC, Cout, bias, hout, Cout, MtG, NtG, Kdp, 1);
        }
    };

    // pipeline
    pool(x, cluster0, 65536, 32768, 1, PBUF);
    conv(PBUF, 32768, 1, 32, ei1, ps1, 524288, W1, r1, b1, HBUF);
    pool(HBUF, cluster1, 32768, 16384, 32, PBUF);
    conv(PBUF, 16384, 32, 64, ei2, ps2, 262144, W2, r2, b2, HBUF);
    pool(HBUF, cluster2, 16384, 8192, 64, PBUF);
    conv(PBUF, 8192, 64, 128, ei3, ps3, 131072, W3, r3, b3, HBUF);
    pool(HBUF, cluster3, 8192, 4096, 128, PBUF);
    conv(PBUF, 4096, 128, 256, ei4, ps4, 65536, W4, r4, b4, HBUF);
    pool(HBUF, cluster4, 4096, 512, 256, PBUF);   // PBUF == [64, 2048] row-major

    // FC1 (WMMA, bias+ELU):  [64,2048] @ [2048,512]
    {
        pack_b_kernel<<<cdiv(2048 * 512, 256), 256, 0, stream>>>(fc1_w, WC, 2048, 512);
        int MtG = 64 / 32, NtG = 512 / 64;
        wmma_gemm_kernel<4><<<cdiv(MtG * NtG, 8), 256, 0, stream>>>(
            PBUF, 2048, WC, 512, fc1_b, HBUF, 512, MtG, NtG, 2048, 1);
    }
    // FC2 + log_softmax
    fc2_logsoftmax_kernel<<<64, 32, 0, stream>>>(HBUF, fc2_w, fc2_b, (float*)d_out);
}